// TargetEdgeEmbedding_57294863729155
// MI455X (gfx1250) — compile-verified
//
#include <hip/hip_runtime.h>
#include <hip/hip_bf16.h>
#include <math.h>

typedef float v2f __attribute__((ext_vector_type(2)));
typedef float v8f __attribute__((ext_vector_type(8)));

#define SQ3    1.7320508075688772f
#define SQ15   3.872983346207417f
#define SQ5H   1.1180339887498949f   // sqrt(5)/2
#define SQ15H  1.9364916731037085f   // sqrt(15)/2
#define INV_SQRT_2PI 0.3989422804014327f

// ---------------------------------------------------------------------------
// Kernel 1: fuse projection + weight-generation matrices.
//   B[r,k]  = sum_v proj[r_seg][u,v] * g[v] * scale_seg * rad_w2[vbase+v, k]
//   c[r]    = sum_v (same a_v) * rad_offset[vbase+v]  (+ proj_b0[r] for seg 0)
// ---------------------------------------------------------------------------
__global__ __launch_bounds__(64)
void prep_kernel(const float* __restrict__ rad_w2,    // (960,64)
                 const float* __restrict__ rad_offset,// (960)
                 const float* __restrict__ exp_w,     // (128)
                 const float* __restrict__ exp_b,     // (128)
                 const float* __restrict__ pw0,       // (128,224)
                 const float* __restrict__ pw1,       // (64,384)
                 const float* __restrict__ pw2,       // (32,352)
                 const float* __restrict__ pb0,       // (128)
                 float* __restrict__ Bmat,            // (224,64)
                 float* __restrict__ cvec)            // (224)
{
    const int r = blockIdx.x;    // 0..223
    const int k = threadIdx.x;   // 0..63
    __shared__ float a[128];

    int vbase; const float* pw; int stride; float scale; int u;
    if (r < 128)      { vbase = 0;   u = r;       pw = pw0; stride = 224; scale = 1.0f / sqrtf(224.0f); }
    else if (r < 192) { vbase = 128; u = r - 128; pw = pw1; stride = 384; scale = 1.0f / sqrtf(384.0f); }
    else              { vbase = 256; u = r - 192; pw = pw2; stride = 352; scale = 1.0f / sqrtf(352.0f); }

    for (int v = k; v < 128; v += 64)
        a[v] = pw[u * stride + v] * (exp_w[v] + exp_b[v]) * scale;
    __syncthreads();

    float s = 0.0f;
    #pragma unroll 8
    for (int v = 0; v < 128; ++v)
        s += a[v] * rad_w2[(vbase + v) * 64 + k];
    Bmat[r * 64 + k] = s;

    if (k == 0) {
        float c = 0.0f;
        for (int v = 0; v < 128; ++v)
            c += a[v] * rad_offset[vbase + v];
        if (r < 128) c += pb0[r];
        cvec[r] = c;
    }
}

// ---------------------------------------------------------------------------
// Kernel 2: fused edge kernel.
//   Stage 1 (lane-per-edge VALU): d, u, sh, RBF, MLP1, LayerNorm, SiLU -> h in LDS
//   Stage 2 (WMMA f32 16x16x4):   per 16-edge tile, D(16x224) = H(16x64) @ B^T,
//                                 then expand with spherical harmonics to 480 cols.
// Block = 128 threads (4 waves, wave32). Grid = E/128.
// n-loop kept rolled (unroll 1) to hold register pressure down -> high occupancy;
// this kernel is memory-bound (63 MB streaming output), so waves-in-flight beat
// static scheduling depth.
// ---------------------------------------------------------------------------
#define HPAD 68   // h row stride in LDS dwords (bank-conflict padding)

__global__ __launch_bounds__(128, 4)
void edge_kernel(const float* __restrict__ X,        // (E,3)
                 const float* __restrict__ rbf_mean, // (32)
                 const float* __restrict__ rbf_std,  // (32)
                 const float* __restrict__ rbf_w,    // (1)
                 const float* __restrict__ rbf_b,    // (1)
                 const float* __restrict__ w1,       // (64,32)
                 const float* __restrict__ ln_g,     // (64)
                 const float* __restrict__ ln_b,     // (64)
                 const float* __restrict__ Bmat,     // (224,64) fused
                 const float* __restrict__ cvec,     // (224)
                 float* __restrict__ out)            // (E,480)
{
    __shared__ float s_mean[32];
    __shared__ float s_std[32];
    __shared__ float s_c[224];
    __shared__ float s_w1[64 * 32];
    __shared__ float s_h[128 * HPAD];
    __shared__ float s_sh[128 * 8];

    const int tid = threadIdx.x;

    // ---- cooperative preload of small params ----
    for (int i = tid; i < 64 * 32; i += 128) s_w1[i] = w1[i];
    if (tid < 32) {
        s_mean[tid] = rbf_mean[tid];
        s_std[tid]  = fabsf(rbf_std[tid]) + 1e-5f;
    }
    for (int i = tid; i < 224; i += 128) s_c[i] = cvec[i];

    // ---- stage 1: one edge per lane ----
    const int e = blockIdx.x * 128 + tid;
    const float x0 = X[3 * e + 0];
    const float x1 = X[3 * e + 1];
    const float x2 = X[3 * e + 2];
    const float d   = sqrtf(x0 * x0 + x1 * x1 + x2 * x2);
    const float inv = 1.0f / d;
    const float ux = x0 * inv, uy = x1 * inv, uz = x2 * inv;

    float* shp = &s_sh[tid * 8];
    shp[0] = SQ3 * uy;
    shp[1] = SQ3 * uz;
    shp[2] = SQ3 * ux;
    shp[3] = SQ15 * ux * uy;
    shp[4] = SQ15 * uy * uz;
    shp[5] = SQ5H * (3.0f * uz * uz - 1.0f);
    shp[6] = SQ15 * ux * uz;
    shp[7] = SQ15H * (ux * ux - uy * uy);

    const float t = rbf_w[0] * (d * 0.2f) + rbf_b[0];
    __syncthreads();   // params + sh visible

    float rbf[32];
    #pragma unroll
    for (int k = 0; k < 32; ++k) {
        const float s = s_std[k];
        const float z = (t - s_mean[k]) / s;
        rbf[k] = expf(-0.5f * z * z) * (INV_SQRT_2PI / s);
    }

    float h[64];
    #pragma unroll 4
    for (int j = 0; j < 64; ++j) {
        float acc = 0.0f;
        #pragma unroll
        for (int k = 0; k < 32; ++k) acc += s_w1[j * 32 + k] * rbf[k];
        h[j] = acc;
    }

    // LayerNorm + SiLU
    float mu = 0.0f;
    #pragma unroll
    for (int j = 0; j < 64; ++j) mu += h[j];
    mu *= (1.0f / 64.0f);
    float var = 0.0f;
    #pragma unroll
    for (int j = 0; j < 64; ++j) { const float dd = h[j] - mu; var += dd * dd; }
    var *= (1.0f / 64.0f);
    const float rstd = rsqrtf(var + 1e-5f);
    #pragma unroll
    for (int j = 0; j < 64; ++j) {
        float hn = (h[j] - mu) * rstd * ln_g[j] + ln_b[j];
        hn = hn / (1.0f + expf(-hn));        // SiLU
        s_h[tid * HPAD + j] = hn;
    }
    __syncthreads();

    // ---- stage 2: WMMA projection, 16-edge tiles ----
    const int lane = tid & 31;
    const int wv   = tid >> 5;
    const int half = lane >> 4;   // 0: K={0,1}/rows 0-7; 1: K={2,3}/rows 8-15
    const int l16  = lane & 15;
    const int koff = half * 2;

    #pragma unroll 1
    for (int tile = 0; tile < 2; ++tile) {
        const int el0  = wv * 32 + tile * 16;   // local edge base of this tile
        const int arow = el0 + l16;

        // A fragments: H[edge, K] in 16x4-f32 layout, 16 K-steps
        v2f afr[16];
        #pragma unroll
        for (int kk = 0; kk < 16; ++kk)
            afr[kk] = *(const v2f*)&s_h[arow * HPAD + 4 * kk + koff];

        #pragma unroll 1
        for (int n = 0; n < 14; ++n) {
            const int bcol = n * 16 + l16;      // output channel of this lane

            v2f bfr[16];
            #pragma unroll
            for (int kk = 0; kk < 16; ++kk)
                bfr[kk] = *(const v2f*)&Bmat[bcol * 64 + 4 * kk + koff];

            const float cval = s_c[bcol];
            v8f acc = { cval, cval, cval, cval, cval, cval, cval, cval };
            #pragma unroll
            for (int kk = 0; kk < 16; ++kk)
                acc = __builtin_amdgcn_wmma_f32_16x16x4_f32(
                        false, afr[kk], false, bfr[kk], (short)0, acc, false, false);

            // scatter to output (C layout: VGPR r -> M = r + 8*half, N = l16)
            #pragma unroll
            for (int r = 0; r < 8; ++r) {
                const int m  = r + half * 8;
                const int el = el0 + m;                       // local edge
                const int eg = blockIdx.x * 128 + el;         // global edge
                const float val = acc[r];
                float* orow = out + eg * 480;
                if (n < 8) {
                    __builtin_nontemporal_store(val, &orow[bcol]);   // l=0 cols
                } else if (n < 12) {
                    const int uidx = bcol - 128;              // l=1 block
                    const float* sp = &s_sh[el * 8];
                    __builtin_nontemporal_store(val * sp[0], &orow[128 + uidx * 3 + 0]);
                    __builtin_nontemporal_store(val * sp[1], &orow[128 + uidx * 3 + 1]);
                    __builtin_nontemporal_store(val * sp[2], &orow[128 + uidx * 3 + 2]);
                } else {
                    const int uidx = bcol - 192;              // l=2 block
                    const float* sp = &s_sh[el * 8 + 3];
                    __builtin_nontemporal_store(val * sp[0], &orow[320 + uidx * 5 + 0]);
                    __builtin_nontemporal_store(val * sp[1], &orow[320 + uidx * 5 + 1]);
                    __builtin_nontemporal_store(val * sp[2], &orow[320 + uidx * 5 + 2]);
                    __builtin_nontemporal_store(val * sp[3], &orow[320 + uidx * 5 + 3]);
                    __builtin_nontemporal_store(val * sp[4], &orow[320 + uidx * 5 + 4]);
                }
            }
        }
    }
}

// ---------------------------------------------------------------------------
extern "C" void kernel_launch(void* const* d_in, const int* in_sizes, int n_in,
                              void* d_out, int out_size, void* d_ws, size_t ws_size,
                              hipStream_t stream) {
    const float* X          = (const float*)d_in[0];
    const float* rbf_mean   = (const float*)d_in[1];
    const float* rbf_std    = (const float*)d_in[2];
    const float* rbf_w      = (const float*)d_in[3];
    const float* rbf_b      = (const float*)d_in[4];
    const float* rad_w1     = (const float*)d_in[5];
    const float* ln_g       = (const float*)d_in[6];
    const float* ln_b       = (const float*)d_in[7];
    const float* rad_w2     = (const float*)d_in[8];
    const float* rad_offset = (const float*)d_in[9];
    const float* exp_w      = (const float*)d_in[10];
    const float* exp_b      = (const float*)d_in[11];
    const float* pw0        = (const float*)d_in[12];
    const float* pw1        = (const float*)d_in[13];
    const float* pw2        = (const float*)d_in[14];
    const float* pb0        = (const float*)d_in[15];

    float* Bmat = (float*)d_ws;          // 224*64 floats
    float* cvec = Bmat + 224 * 64;       // 224 floats

    const int E = in_sizes[0] / 3;       // 32768

    prep_kernel<<<224, 64, 0, stream>>>(rad_w2, rad_offset, exp_w, exp_b,
                                        pw0, pw1, pw2, pb0, Bmat, cvec);

    edge_kernel<<<E / 128, 128, 0, stream>>>(X, rbf_mean, rbf_std, rbf_w, rbf_b,
                                             rad_w1, ln_g, ln_b, Bmat, cvec,
                                             (float*)d_out);
}